// MessagePassAttention_50173807952913
// MI455X (gfx1250) — compile-verified
//
#include <hip/hip_runtime.h>
#include <stdint.h>
#include <math.h>

// ---------------------------------------------------------------------------
// MessagePassAttention on MI455X (gfx1250, wave32, WMMA + TDM).
//
// Per edge (one wave32):
//   TDM: tensor_load_to_lds streams M (36x36 f32, 5184B contiguous) -> LDS
//   Mek|Mev = M(36x36) @ [ek|ev](36x8)   -> 27x v_wmma_f32_16x16x4_f32
//   alpha   = eq^T @ Mek (4x4)           -> per-lane VALU via LDS
//   alpha   = MLP16(silu) (16x16 x2)     -> per-lane VALU
//   out     = Mev @ alpha (36x4)         -> 3x v_wmma_f32_16x16x4_f32
//   message[src] += out                  -> global_atomic_add_f32 (L2-resident)
//
// Roofline: 415 MB of edge matrices streamed once -> ~17.8us floor @23.3TB/s;
// q/k/v gathers (8.6MB) and the atomic output (2.9MB) are L2-resident.
// Padding is materialized as zeros in LDS so every WMMA fragment load is
// unconditional and multiply-free (EXEC stays full, no select VALU).
// ---------------------------------------------------------------------------

typedef float v2f __attribute__((ext_vector_type(2)));
typedef float v8f __attribute__((ext_vector_type(8)));
typedef unsigned int v4u __attribute__((ext_vector_type(4)));
typedef int v4i __attribute__((ext_vector_type(4)));
typedef int v8i __attribute__((ext_vector_type(8)));

#define N_NODES  5000
#define BASIS    36
#define CHAN     4
#define N_EDGES  80000
#define EDGES_PER_BLOCK 8          // 8 waves * 32 lanes = 256 threads
#define M_ELEMS  (BASIS * BASIS)   // 1296
#define M_PAD    (48 * BASIS)      // 48 rows x 36 cols (rows 36..47 zero)

__global__ void mpa_zero_kernel(float* __restrict__ out, int n) {
  int i = blockIdx.x * blockDim.x + threadIdx.x;
  if (i < n) out[i] = 0.0f;
}

__global__ __launch_bounds__(256)
void mpa_edge_kernel(const float* __restrict__ query,
                     const float* __restrict__ key,
                     const float* __restrict__ value,
                     const float* __restrict__ edge_mat,
                     const int*   __restrict__ edge_index,
                     const float* __restrict__ W1, const float* __restrict__ b1,
                     const float* __restrict__ W2, const float* __restrict__ b2,
                     float* __restrict__ out)
{
  // Per-wave scratch (8 waves per block)
  __shared__ float Mlds[EDGES_PER_BLOCK][M_PAD];        // 48x36, TDM fills rows 0..35
  __shared__ float S[EDGES_PER_BLOCK][48][8];           // [Mek | Mev], rows 36..47 = 0
  __shared__ float kvS[EDGES_PER_BLOCK][BASIS * 16];    // RHS 36x16: [k|v|zeros]
  __shared__ float eqS[EDGES_PER_BLOCK][BASIS * CHAN];  // gathered q[src]
  __shared__ float hS[EDGES_PER_BLOCK][16];
  __shared__ float sS[EDGES_PER_BLOCK][16];
  __shared__ float aS[EDGES_PER_BLOCK][64];             // alpha as 4x16 (cols 4..15 zeroed in rows 0..1 only)
  __shared__ float W1s[256], W2s[256], b1s[16], b2s[16];

  const int tid  = threadIdx.x;
  const int w    = tid >> 5;        // wave id in block
  const int lane = tid & 31;

  // Stage MLP weights once per block
  W1s[tid] = W1[tid];
  W2s[tid] = W2[tid];
  if (tid < 16) { b1s[tid] = b1[tid]; b2s[tid] = b2[tid]; }

  const int eraw    = blockIdx.x * EDGES_PER_BLOCK + w;
  const bool active = (eraw < N_EDGES);
  const int e       = active ? eraw : (N_EDGES - 1);

  int src = edge_index[e];
  int dst = edge_index[N_EDGES + e];
  src = __builtin_amdgcn_readfirstlane(src);   // wave-uniform
  dst = __builtin_amdgcn_readfirstlane(dst);

  // ---- Kick off TDM: stream M[e] (contiguous 5184B) into this wave's LDS ---
  const float* __restrict__ Mg = edge_mat + (size_t)e * M_ELEMS;
#if defined(__has_builtin) && __has_builtin(__builtin_amdgcn_tensor_load_to_lds)
  {
    uint32_t ldsoff = (uint32_t)(uintptr_t)(&Mlds[w][0]);        // LDS byte offset
    ldsoff = (uint32_t)__builtin_amdgcn_readfirstlane((int)ldsoff);
    uint64_t ga = (uint64_t)(uintptr_t)Mg;
    uint32_t galo = (uint32_t)__builtin_amdgcn_readfirstlane((int)(uint32_t)ga);
    uint32_t gahi = (uint32_t)__builtin_amdgcn_readfirstlane((int)(uint32_t)(ga >> 32));
    // D# group0: count=1 | lds_addr[63:32] | global_addr[120:64] | type=2
    v4u g0 = { 1u, ldsoff, galo, (gahi & 0x01FFFFFFu) | (2u << 30) };
    // D# group1: data_size=4B; tensor_dim0 = tile_dim0 = 1296; 1D copy
    v8i g1 = { 0x00020000,                       // data_size=2 (4B), mask=0
               (int)((M_ELEMS & 0xFFFF) << 16),  // tensor_dim0[15:0] @ [63:48]
               0,                                // tensor_dim0[31:16], tensor_dim1 lo
               (int)((M_ELEMS & 0xFFFF) << 16),  // tile_dim0 @ [127:112]
               0,                                // tile_dim1=0, tile_dim2=0 (unused)
               M_ELEMS,                          // tensor_dim0_stride lo
               0, 0 };
    v4i gz = { 0, 0, 0, 0 };
#if __clang_major__ >= 23
    v8i gz8 = { 0, 0, 0, 0, 0, 0, 0, 0 };
    __builtin_amdgcn_tensor_load_to_lds(g0, g1, gz, gz, gz8, 0);
#else
    __builtin_amdgcn_tensor_load_to_lds(g0, g1, gz, gz, 0);
#endif
  }
#else
  // Fallback: coalesced NT loads into LDS
  for (int i = lane; i < M_ELEMS; i += 32)
    Mlds[w][i] = __builtin_nontemporal_load(Mg + i);
#endif

  // ---- Overlap with TDM: zero pads, stage RHS (36x16) and eq (36x4) --------
  for (int i = lane; i < M_PAD - M_ELEMS; i += 32)       // zero pad rows 36..47
    Mlds[w][M_ELEMS + i] = 0.0f;
  if (lane < 16) aS[w][lane * 2] = 0.0f, aS[w][lane * 2 + 1] = 0.0f;
  for (int i = lane; i < BASIS * 16; i += 32) {          // RHS[k][n], zero n>=8
    const int k = i >> 4, n = i & 15;
    float v = 0.0f;
    if (n < 4)      v = key  [((size_t)dst * BASIS + k) * CHAN + n];
    else if (n < 8) v = value[((size_t)dst * BASIS + k) * CHAN + (n - 4)];
    kvS[w][i] = v;
  }
  for (int i = lane; i < BASIS * CHAN; i += 32)
    eqS[w][i] = query[(size_t)src * (BASIS * CHAN) + i];

  // Wait for this wave's tensor DMA, then make LDS staging visible block-wide
#if defined(__has_builtin) && __has_builtin(__builtin_amdgcn_s_wait_tensorcnt)
  __builtin_amdgcn_s_wait_tensorcnt(0);
#endif
  asm volatile("" ::: "memory");
  __syncthreads();

  // ---- Big GEMM: [Mek|Mev](48x16) = Mpad(48x36) @ RHS(36x16) ---------------
  // All fragment loads are unconditional (zero padding lives in LDS).
  const float* __restrict__ Mw = &Mlds[w][0];
  const int rlo  = lane & 15;            // A-frag row within tile / C col
  const int koff = (lane < 16) ? 0 : 2;  // K pair held by this lane half
  const int ncol = lane & 15;            // B-frag column

  v8f ct[3];
  #pragma unroll
  for (int t = 0; t < 3; ++t)
    ct[t] = (v8f){0.f, 0.f, 0.f, 0.f, 0.f, 0.f, 0.f, 0.f};

  #pragma unroll
  for (int kt = 0; kt < 9; ++kt) {       // K = 36 = 9 * 4
    const int k0 = kt * 4 + koff;
    v2f bfrag;
    bfrag.x = kvS[w][k0 * 16 + ncol];
    bfrag.y = kvS[w][(k0 + 1) * 16 + ncol];

    #pragma unroll
    for (int t = 0; t < 3; ++t) {
      const int row = t * 16 + rlo;
      v2f afrag;                          // ds_load_b64 pair along M row
      afrag.x = Mw[row * BASIS + kt * 4 + koff];
      afrag.y = Mw[row * BASIS + kt * 4 + koff + 1];
      ct[t] = __builtin_amdgcn_wmma_f32_16x16x4_f32(
                  false, afrag, false, bfrag, (short)0, ct[t], false, false);
    }
  }

  // Spill C tiles (only cols 0..7 meaningful) to LDS; rows 36..47 are zeros.
  {
    const int col = lane & 15;
    #pragma unroll
    for (int t = 0; t < 3; ++t) {
      const int rbase = t * 16 + ((lane < 16) ? 0 : 8);
      if (col < 8) {
        #pragma unroll
        for (int v = 0; v < 8; ++v)
          S[w][rbase + v][col] = ct[t][v];
      }
    }
  }
  __syncthreads();

  // ---- alpha[c][m] = sum_i eq[i][c] * Mek[i][m]  (lanes 0..15, lane=c*4+m) -
  if (lane < 16) {
    const int c = lane >> 2, m = lane & 3;
    float acc = 0.0f;
    for (int i = 0; i < BASIS; ++i)
      acc += eqS[w][i * CHAN + c] * S[w][i][m];
    hS[w][lane] = acc;
  }
  __syncthreads();

  // ---- MLP: silu(h @ W1^T + b1) @ W2^T + b2 --------------------------------
  if (lane < 16) {
    float acc = b1s[lane];
    #pragma unroll
    for (int k = 0; k < 16; ++k) acc += hS[w][k] * W1s[lane * 16 + k];
    sS[w][lane] = acc / (1.0f + __expf(-acc));   // silu
  }
  __syncthreads();
  if (lane < 16) {
    float acc = b2s[lane];
    #pragma unroll
    for (int k = 0; k < 16; ++k) acc += sS[w][k] * W2s[lane * 16 + k];
    aS[w][(lane >> 2) * 16 + (lane & 3)] = acc;  // alpha[c][m] into 4x16 (padded)
  }
  __syncthreads();

  // ---- out(36x4) = Mev(36x4) @ alpha(4x4): 3 WMMAs, K = 4 ------------------
  v2f bfin;                               // B[k][n] = aS[k*16+n] (n>=4 feeds only discarded cols)
  bfin.x = aS[w][koff * 16 + ncol];
  bfin.y = aS[w][(koff + 1) * 16 + ncol];

  v8f ot[3];
  #pragma unroll
  for (int t = 0; t < 3; ++t) {
    const int row = t * 16 + rlo;         // S rows 36..47 hold zeros
    v2f afrag;
    afrag.x = S[w][row][4 + koff];        // Mev channel koff
    afrag.y = S[w][row][4 + koff + 1];    // Mev channel koff+1
    v8f z = (v8f){0.f, 0.f, 0.f, 0.f, 0.f, 0.f, 0.f, 0.f};
    ot[t] = __builtin_amdgcn_wmma_f32_16x16x4_f32(
                false, afrag, false, bfin, (short)0, z, false, false);
  }

  // ---- scatter-add into message[src] (L2-resident fp32 atomics) ------------
  if (active && ncol < 4) {
    #pragma unroll
    for (int t = 0; t < 3; ++t) {
      const int rbase = t * 16 + ((lane < 16) ? 0 : 8);
      #pragma unroll
      for (int v = 0; v < 8; ++v) {
        const int row = rbase + v;
        if (row < BASIS) {
          float* p = out + ((size_t)src * BASIS + row) * CHAN + ncol;
          float val = ot[t][v];
          asm volatile("global_atomic_add_f32 %0, %1, off"
                       :: "v"(p), "v"(val) : "memory");
        }
      }
    }
  }
}

extern "C" void kernel_launch(void* const* d_in, const int* in_sizes, int n_in,
                              void* d_out, int out_size, void* d_ws, size_t ws_size,
                              hipStream_t stream) {
  const float* query    = (const float*)d_in[0];
  const float* key      = (const float*)d_in[1];
  const float* value    = (const float*)d_in[2];
  const float* edge_mat = (const float*)d_in[3];
  const int*   eidx     = (const int*)  d_in[4];
  // d_in[5] = coeff_ind_to_node_ind: arange(T)//BASIS -> implicit in indexing
  const float* W1 = (const float*)d_in[6];
  const float* b1 = (const float*)d_in[7];
  const float* W2 = (const float*)d_in[8];
  const float* b2 = (const float*)d_in[9];
  float* out = (float*)d_out;

  // We accumulate with atomics -> zero the output every call
  mpa_zero_kernel<<<(out_size + 255) / 256, 256, 0, stream>>>(out, out_size);

  int nblocks = (N_EDGES + EDGES_PER_BLOCK - 1) / EDGES_PER_BLOCK;  // 10000
  mpa_edge_kernel<<<nblocks, 256, 0, stream>>>(query, key, value, edge_mat,
                                               eidx, W1, b1, W2, b2, out);
}